// RNNNetv1_240518169257
// MI455X (gfx1250) — compile-verified
//
#include <hip/hip_runtime.h>
#include <math.h>

typedef __attribute__((ext_vector_type(2))) float v2f;
typedef __attribute__((ext_vector_type(8))) float v8f;

#define Bz 10
#define Tz 12
#define Fz 500
#define Hz 200
#define G3 600      // 3*H
#define CTXS 512    // padded context row stride
#define GS 608      // padded gate row stride
#define NT 38       // ceil(600/16) N-tiles
#define NTHREADS 512

extern __shared__ float lds[];

__launch_bounds__(NTHREADS, 1)
__global__ void rnn_gru_attn_fused(const float* __restrict__ x,
                                   const float* __restrict__ guidance,
                                   const float* __restrict__ wx,
                                   const float* __restrict__ Wh,
                                   const float* __restrict__ bh,
                                   const float* __restrict__ W_ih,
                                   const float* __restrict__ W_hh,
                                   const float* __restrict__ b_ih,
                                   const float* __restrict__ b_hh,
                                   const float* __restrict__ h0,
                                   float* __restrict__ out)
{
    // LDS carve-up (~135 KB total)
    float* xw    = lds;                 // 120  : x[b,t]·wx (loop invariant)
    float* gmask = xw + 120;            // 120  : guidance mask per step
    float* s     = gmask + 120;         // 120  : attention scores
    float* a     = s + 120;             // 120  : softmax weights
    float* hA    = a + 120;             // 16*200 (rows 10..15 stay zero)
    float* hB    = hA + 16 * Hz;        // 16*200
    float* ctx   = hB + 16 * Hz;        // 16*512 (rows 10..15 / cols 500..511 stay zero)
    float* gi    = ctx + 16 * CTXS;     // 16*608
    float* gh    = gi + 16 * GS;        // 16*608

    const int tid = threadIdx.x;

    // ---- one-time init ----
    for (int idx = tid; idx < 16 * Hz; idx += NTHREADS) { hA[idx] = 0.f; hB[idx] = 0.f; }
    for (int idx = tid; idx < 16 * CTXS; idx += NTHREADS) ctx[idx] = 0.f;
    for (int idx = tid; idx < Bz * Hz; idx += NTHREADS) {
        int b = idx / Hz, j = idx % Hz;
        hA[b * Hz + j] = h0[idx];
    }
    // xw[b][t] = dot(x[b,t,:], wx) — invariant across steps
    if (tid < Bz * Tz) {
        const float* xp = x + tid * Fz;
        float acc = 0.f;
        for (int f = 0; f < Fz; ++f) acc += xp[f] * wx[f];
        xw[tid] = acc;
    }
    __syncthreads();

    const int lane = tid & 31;
    const int wave = tid >> 5;
    const int mrow = lane & 15;          // M (or N for the B operand)
    const int khi  = (lane >> 4) * 2;    // K sub-offset per lane half
    const int mbase = (lane >> 4) * 8;   // D-matrix M base

    for (int i = 0; i < Tz; ++i) {
        float* hcur = (i & 1) ? hB : hA;
        float* hnxt = (i & 1) ? hA : hB;

        // P1: guidance mask g[b][t], t<=i
        if (tid < Bz) {
            int b = tid;
            float gsum = 0.f;
            for (int t = 0; t < i; ++t) gsum += guidance[b * Tz + t];
            bool zrow = (gsum == 0.f);
            for (int t = 0; t < i; ++t)
                gmask[b * Tz + t] = zrow ? 1.f : guidance[b * Tz + t];
            gmask[b * Tz + i] = 1.f;
        }
        __syncthreads();

        // P2: scores s[b][t] = g*xw + h·Wh[t] + bh[t], for t<=i
        if (tid < Bz * Tz) {
            int b = tid / Tz, t = tid % Tz;
            if (t <= i) {
                float acc = bh[t];
                const float* wrow = Wh + t * Hz;
                const float* hrow = hcur + b * Hz;
                for (int k = 0; k < Hz; ++k) acc += hrow[k] * wrow[k];
                s[tid] = gmask[tid] * xw[tid] + acc;
            }
        }
        __syncthreads();

        // P3: softmax over the BATCH axis (legacy implicit dim=0), per t<=i
        if (tid <= i) {
            int t = tid;
            float m = -1e30f;
            for (int b = 0; b < Bz; ++b) m = fmaxf(m, s[b * Tz + t]);
            float sum = 0.f;
            float e[Bz];
            for (int b = 0; b < Bz; ++b) { e[b] = expf(s[b * Tz + t] - m); sum += e[b]; }
            float inv = 1.f / sum;
            for (int b = 0; b < Bz; ++b) a[b * Tz + t] = e[b] * inv;
        }
        __syncthreads();

        // P4: context[b][f] = sum_{t<=i} x[b,t,f] * a[b,t]
        for (int idx = tid; idx < Bz * Fz; idx += NTHREADS) {
            int b = idx / Fz, f = idx % Fz;
            const float* xp = x + b * Tz * Fz + f;
            float acc = 0.f;
            for (int t = 0; t <= i; ++t) acc += xp[t * Fz] * a[b * Tz + t];
            ctx[b * CTXS + f] = acc;
        }
        __syncthreads();

        // P5: WMMA f32 GEMMs: gi = ctx @ W_ih^T (K=500), gh = h @ W_hh^T (K=200)
        for (int tile = wave; tile < NT; tile += (NTHREADS / 32)) {
            int j  = tile * 16 + mrow;               // N column (weight row)
            int jc = j < G3 ? j : (G3 - 1);          // clamped, branchless
            float jm = j < G3 ? 1.f : 0.f;           // zero-mask for padded columns

            const float* wrow = W_ih + jc * Fz;
            const float* arow = ctx + mrow * CTXS;
            v8f accI = {};
            for (int c = 0; c < Fz / 4; ++c) {
                int base = 4 * c + khi;
                v2f av = { arow[base], arow[base + 1] };
                v2f bv = { wrow[base] * jm, wrow[base + 1] * jm };
                accI = __builtin_amdgcn_wmma_f32_16x16x4_f32(
                    false, av, false, bv, (short)0, accI, false, false);
            }

            const float* wrow2 = W_hh + jc * Hz;
            const float* hrow  = hcur + mrow * Hz;
            v8f accH = {};
            for (int c = 0; c < Hz / 4; ++c) {
                int base = 4 * c + khi;
                v2f av = { hrow[base], hrow[base + 1] };
                v2f bv = { wrow2[base] * jm, wrow2[base + 1] * jm };
                accH = __builtin_amdgcn_wmma_f32_16x16x4_f32(
                    false, av, false, bv, (short)0, accH, false, false);
            }

            // D layout: VGPR v -> M = mbase+v, N = lane&15 within tile
            int n = tile * 16 + mrow;
            if (n < G3) {
                #pragma unroll
                for (int v = 0; v < 8; ++v) {
                    gi[(mbase + v) * GS + n] = accI[v];
                    gh[(mbase + v) * GS + n] = accH[v];
                }
            }
        }
        __syncthreads();

        // P6: GRU gate math + hidden update
        for (int idx = tid; idx < Bz * Hz; idx += NTHREADS) {
            int b = idx / Hz, j = idx % Hz;
            const float* gib = gi + b * GS;
            const float* ghb = gh + b * GS;
            float ir  = gib[j]          + b_ih[j];
            float hr  = ghb[j]          + b_hh[j];
            float iz  = gib[j + Hz]     + b_ih[j + Hz];
            float hz  = ghb[j + Hz]     + b_hh[j + Hz];
            float in_ = gib[j + 2 * Hz] + b_ih[j + 2 * Hz];
            float hn  = ghb[j + 2 * Hz] + b_hh[j + 2 * Hz];
            float r = 1.f / (1.f + expf(-(ir + hr)));
            float z = 1.f / (1.f + expf(-(iz + hz)));
            float n = tanhf(in_ + r * hn);
            hnxt[b * Hz + j] = (1.f - z) * n + z * hcur[b * Hz + j];
        }
        __syncthreads();
    }

    // 12 steps (even) -> final h lives in hA
    for (int idx = tid; idx < Bz * Hz; idx += NTHREADS) {
        out[idx] = hA[idx / Hz * Hz + idx % Hz];
    }
}

extern "C" void kernel_launch(void* const* d_in, const int* in_sizes, int n_in,
                              void* d_out, int out_size, void* d_ws, size_t ws_size,
                              hipStream_t stream) {
    (void)in_sizes; (void)n_in; (void)out_size; (void)d_ws; (void)ws_size;
    const float* x        = (const float*)d_in[0];
    const float* guidance = (const float*)d_in[1];
    const float* wx       = (const float*)d_in[2];
    const float* Wh       = (const float*)d_in[3];
    const float* bh       = (const float*)d_in[4];
    const float* W_ih     = (const float*)d_in[5];
    const float* W_hh     = (const float*)d_in[6];
    const float* b_ih     = (const float*)d_in[7];
    const float* b_hh     = (const float*)d_in[8];
    const float* h0       = (const float*)d_in[9];
    float* out = (float*)d_out;

    // LDS bytes: (120*4 + 2*16*200 + 16*512 + 2*16*608) floats
    size_t shmem = (size_t)(120 * 4 + 2 * 16 * Hz + 16 * CTXS + 2 * 16 * GS) * sizeof(float);

    rnn_gru_attn_fused<<<1, NTHREADS, shmem, stream>>>(
        x, guidance, wx, Wh, bh, W_ih, W_hh, b_ih, b_hh, h0, out);
}